// EfficientAttention_17334488006918
// MI455X (gfx1250) — compile-verified
//
#include <hip/hip_runtime.h>
#include <hip/hip_bf16.h>

typedef __attribute__((ext_vector_type(2))) float v2f;
typedef __attribute__((ext_vector_type(4))) float v4f;
typedef __attribute__((ext_vector_type(8))) float v8f;

#define N_PER   262144      // 64*64*64 tokens per batch image
#define NC      64          // channels
#define NH      4           // heads
#define DKH     16          // channels per head
#define TILE    128         // tokens per workgroup
#define THREADS 256         // 8 waves (wave32)
#define XPAD    68          // LDS row stride (floats); bank-clean + pipeline pad
#define CPAD    17          // LDS row stride for ctx 16x16 blocks
#define NSLOT   64          // partial-accumulation slots (atomic contention / 64)

// D(16x16,f32) = A(16x4,f32) * B(4x16,f32) + C  — CDNA5 fp32 WMMA
__device__ __forceinline__ v8f wmma4(v2f a, v2f b, v8f c) {
  return __builtin_amdgcn_wmma_f32_16x16x4_f32(
      false, a, false, b, (short)0, c, false, false);
}
__device__ __forceinline__ v8f vzero8() {
  v8f z = {0.f,0.f,0.f,0.f,0.f,0.f,0.f,0.f};
  return z;
}

// ---------------------------------------------------------------------------
// Pass 1: k = x@Wk^T+bk, v = x@Wv^T+bv (WMMA); per-WG partials of
//   se_part[slot][n][c]        = sum exp(k[c])
//   ctx_part[slot][n][h,kc,vc] = sum exp(k[h,kc]) * v[h,vc]
// ---------------------------------------------------------------------------
__global__ __launch_bounds__(THREADS) void ea_pass1(
    const float* __restrict__ x,
    const float* __restrict__ Wk, const float* __restrict__ bk,
    const float* __restrict__ Wv, const float* __restrict__ bv,
    float* __restrict__ ctx_part, float* __restrict__ se_part)
{
  extern __shared__ float sm[];
  float* xs  = sm;                 // TILE x XPAD   (x tile, later exp(k))
  float* wkL = xs  + TILE*XPAD;    // 64 x XPAD
  float* wvL = wkL + 64*XPAD;
  float* vsL = wvL + 64*XPAD;      // TILE x XPAD   (v tile)
  float* bkL = vsL + TILE*XPAD;    // 64
  float* bvL = bkL + 64;           // 64
  float* seL = bvL + 64;           // 64 (per-WG sumexp accumulator)

  const int tid  = threadIdx.x;
  const long t0  = (long)blockIdx.x * TILE;
  const int  nb  = (int)(t0 / N_PER);
  const int  slot= blockIdx.x & (NSLOT - 1);

  if (tid < 64) { bkL[tid] = bk[tid]; bvL[tid] = bv[tid]; seL[tid] = 0.f; }

  const float4* wk4 = (const float4*)Wk;
  const float4* wv4 = (const float4*)Wv;
  for (int i = tid; i < 64*16; i += THREADS) {
    int r = i >> 4, c4 = (i & 15) << 2;
    float4 a = wk4[i], b = wv4[i];
    float* pk = &wkL[r*XPAD + c4];
    float* pv = &wvL[r*XPAD + c4];
    pk[0]=a.x; pk[1]=a.y; pk[2]=a.z; pk[3]=a.w;
    pv[0]=b.x; pv[1]=b.y; pv[2]=b.z; pv[3]=b.w;
  }
  const float4* x4 = (const float4*)(x + t0*NC);
  for (int i = tid; i < TILE*16; i += THREADS) {
    int r = i >> 4, c4 = (i & 15) << 2;
    float4 a = x4[i];
    float* p = &xs[r*XPAD + c4];
    p[0]=a.x; p[1]=a.y; p[2]=a.z; p[3]=a.w;
  }
  if (t0 + TILE < 2L*N_PER)                       // prefetch next x tile (32KB)
    __builtin_prefetch(x + (t0 + TILE)*NC + tid*32, 0, 1);
  __syncthreads();

  const int wave = tid >> 5;
  const int lane = tid & 31;
  const int ln = lane & 15, half = lane >> 4;

  v8f ck[4] = {vzero8(), vzero8(), vzero8(), vzero8()};
  v8f cv[4] = {vzero8(), vzero8(), vzero8(), vzero8()};

  // software-pipelined k/v projection: fragments loaded one K-chunk ahead.
  // kc==16 over-reads land in the 4-float row pad (in-bounds, never used).
  const float* arow = &xs[(wave*16 + ln)*XPAD + 2*half];
  const float* krow = &wkL[ln*XPAD + 2*half];
  const float* vrow = &wvL[ln*XPAD + 2*half];

  v2f a_c = *(const v2f*)(arow);
  v2f fk_c[4], fv_c[4];
  #pragma unroll
  for (int cg = 0; cg < 4; ++cg) {
    fk_c[cg] = *(const v2f*)(krow + cg*16*XPAD);
    fv_c[cg] = *(const v2f*)(vrow + cg*16*XPAD);
  }
  #pragma unroll
  for (int kc = 0; kc < 16; ++kc) {
    v2f a_n = *(const v2f*)(arow + (kc+1)*4);
    v2f fk_n[4], fv_n[4];
    #pragma unroll
    for (int cg = 0; cg < 4; ++cg) {
      fk_n[cg] = *(const v2f*)(krow + cg*16*XPAD + (kc+1)*4);
      fv_n[cg] = *(const v2f*)(vrow + cg*16*XPAD + (kc+1)*4);
    }
    #pragma unroll
    for (int cg = 0; cg < 4; ++cg) {
      ck[cg] = wmma4(a_c, fk_c[cg], ck[cg]);
      cv[cg] = wmma4(a_c, fv_c[cg], cv[cg]);
    }
    a_c = a_n;
    #pragma unroll
    for (int cg = 0; cg < 4; ++cg) { fk_c[cg] = fk_n[cg]; fv_c[cg] = fv_n[cg]; }
  }

  // bias + exp(k); stage exp(k) over xs, v into vsL; register-side sumexp
  #pragma unroll
  for (int cg = 0; cg < 4; ++cg) {
    float bbk = bkL[cg*16+ln];
    float bbv = bvL[cg*16+ln];
    float se  = 0.f;
    #pragma unroll
    for (int i = 0; i < 8; ++i) {
      int row = wave*16 + i + 8*half;        // C/D layout: VGPR i -> rows i, i+8
      float e = __expf(ck[cg][i] + bbk);
      se += e;
      xs [row*XPAD + cg*16 + ln] = e;
      vsL[row*XPAD + cg*16 + ln] = cv[cg][i] + bbv;
    }
    se += __shfl_xor(se, 16, 32);            // merge halves: 16-token channel sum
    if (half == 0) atomicAdd(&seL[cg*16 + ln], se);   // ds_add_f32
  }
  __syncthreads();

  // partial ctx: thread = (h, kc, vc-group-of-4); 2 LDS ops per token
  {
    const int vcg = (tid & 3) << 2;
    const int kc  = (tid >> 2) & 15;
    const int h   = tid >> 6;
    v4f acc = {0.f, 0.f, 0.f, 0.f};
    #pragma unroll 4
    for (int t = 0; t < TILE; ++t) {
      float e  = xs[t*XPAD + h*16 + kc];
      v4f   vv = *(const v4f*)&vsL[t*XPAD + h*16 + vcg];
      acc.x += e*vv.x; acc.y += e*vv.y; acc.z += e*vv.z; acc.w += e*vv.w;
    }
    float* cp = &ctx_part[(size_t)slot*2048 + nb*1024 + (h*16 + kc)*16 + vcg];
    atomicAdd(cp+0, acc.x); atomicAdd(cp+1, acc.y);
    atomicAdd(cp+2, acc.z); atomicAdd(cp+3, acc.w);
  }
  if (tid < 64)
    atomicAdd(&se_part[(size_t)slot*128 + nb*64 + tid], seL[tid]);
}

// ---------------------------------------------------------------------------
// Reduce slots + normalize: ctxn[n][h][vc][kc] (transposed for pass-2 B frags)
// ---------------------------------------------------------------------------
__global__ void ea_norm(const float* __restrict__ ctx_part,
                        const float* __restrict__ se_part,
                        float* __restrict__ ctxn)
{
  int i = blockIdx.x * blockDim.x + threadIdx.x;   // cell in [n][h][kc][vc]
  if (i >= 2*NH*DKH*DKH) return;
  int vc = i & 15, kc = (i >> 4) & 15, h = (i >> 8) & 3, nb = i >> 10;
  float c = 0.f, se = 0.f;
  int ch = nb*NC + h*DKH + kc;
  #pragma unroll 4
  for (int s = 0; s < NSLOT; ++s) {
    c  += ctx_part[(size_t)s*2048 + i];
    se += se_part [(size_t)s*128  + ch];
  }
  ctxn[((nb*NH + h)*DKH + vc)*DKH + kc] = c / se;
}

// ---------------------------------------------------------------------------
// Pass 2: q-proj (WMMA) -> head-channel softmax (shfl_xor) ->
//         att = q_soft @ ctx (block-diag WMMA) -> out = att @ Wr^T + br (WMMA)
//         -> staged b128 non-temporal stores.
// ---------------------------------------------------------------------------
__global__ __launch_bounds__(THREADS) void ea_pass2(
    const float* __restrict__ x,
    const float* __restrict__ Wq, const float* __restrict__ bq,
    const float* __restrict__ Wr, const float* __restrict__ br,
    const float* __restrict__ ctxn, float* __restrict__ out)
{
  extern __shared__ float sm[];
  float* xs   = sm;                  // TILE x XPAD
  float* wqL  = xs   + TILE*XPAD;    // 64 x XPAD
  float* wrL  = wqL  + 64*XPAD;      // 64 x XPAD
  float* sL   = wrL  + 64*XPAD;      // TILE x XPAD (q_soft -> att -> out)
  float* ctxL = sL   + TILE*XPAD;    // 4*16*CPAD   ([h][vc][kc])
  float* bqL  = ctxL + NH*DKH*CPAD;  // 64
  float* brL  = bqL  + 64;           // 64

  const int tid = threadIdx.x;
  const long t0 = (long)blockIdx.x * TILE;
  const int  nb = (int)(t0 / N_PER);

  const float4* wq4 = (const float4*)Wq;
  const float4* wr4 = (const float4*)Wr;
  for (int i = tid; i < 64*16; i += THREADS) {
    int r = i >> 4, c4 = (i & 15) << 2;
    float4 a = wq4[i], b = wr4[i];
    float* pq = &wqL[r*XPAD + c4];
    float* pr = &wrL[r*XPAD + c4];
    pq[0]=a.x; pq[1]=a.y; pq[2]=a.z; pq[3]=a.w;
    pr[0]=b.x; pr[1]=b.y; pr[2]=b.z; pr[3]=b.w;
  }
  if (tid < 64) { bqL[tid] = bq[tid]; brL[tid] = br[tid]; }
  for (int i = tid; i < NH*DKH*DKH; i += THREADS) {
    int h = i >> 8, vv = (i >> 4) & 15, kk = i & 15;
    ctxL[(h*DKH + vv)*CPAD + kk] = ctxn[nb*NH*DKH*DKH + i];
  }
  const float4* x4 = (const float4*)(x + t0*NC);
  for (int i = tid; i < TILE*16; i += THREADS) {
    int r = i >> 4, c4 = (i & 15) << 2;
    float4 a = x4[i];
    float* p = &xs[r*XPAD + c4];
    p[0]=a.x; p[1]=a.y; p[2]=a.z; p[3]=a.w;
  }
  if (t0 + TILE < 2L*N_PER)
    __builtin_prefetch(x + (t0 + TILE)*NC + tid*32, 0, 1);
  __syncthreads();

  const int wave = tid >> 5;
  const int lane = tid & 31;
  const int ln = lane & 15, half = lane >> 4;

  // ---- 1) q projection (pipelined) ----
  v8f cq[4] = {vzero8(), vzero8(), vzero8(), vzero8()};
  const float* arow = &xs[(wave*16 + ln)*XPAD + 2*half];
  const float* qrow = &wqL[ln*XPAD + 2*half];
  {
    v2f a_c = *(const v2f*)(arow);
    v2f fb_c[4];
    #pragma unroll
    for (int cg = 0; cg < 4; ++cg)
      fb_c[cg] = *(const v2f*)(qrow + cg*16*XPAD);
    #pragma unroll
    for (int kc = 0; kc < 16; ++kc) {
      v2f a_n = *(const v2f*)(arow + (kc+1)*4);
      v2f fb_n[4];
      #pragma unroll
      for (int cg = 0; cg < 4; ++cg)
        fb_n[cg] = *(const v2f*)(qrow + cg*16*XPAD + (kc+1)*4);
      #pragma unroll
      for (int cg = 0; cg < 4; ++cg)
        cq[cg] = wmma4(a_c, fb_c[cg], cq[cg]);
      a_c = a_n;
      #pragma unroll
      for (int cg = 0; cg < 4; ++cg) fb_c[cg] = fb_n[cg];
    }
  }

  // ---- 2) softmax over 16 head channels (within 16-lane halves) ----
  #pragma unroll
  for (int cg = 0; cg < 4; ++cg) {
    float bb = bqL[cg*16+ln];
    #pragma unroll
    for (int i = 0; i < 8; ++i) {
      float e = __expf(cq[cg][i] + bb);
      float s = e;
      s += __shfl_xor(s, 1, 32);
      s += __shfl_xor(s, 2, 32);
      s += __shfl_xor(s, 4, 32);
      s += __shfl_xor(s, 8, 32);
      cq[cg][i] = e / s;
    }
  }
  #pragma unroll
  for (int cg = 0; cg < 4; ++cg)
    #pragma unroll
    for (int i = 0; i < 8; ++i)
      sL[(wave*16 + i + 8*half)*XPAD + cg*16 + ln] = cq[cg][i];

  // ---- 3) att = q_soft @ ctx (block-diagonal per head) ----
  v8f ca[4];
  #pragma unroll
  for (int h = 0; h < 4; ++h) {
    ca[h] = vzero8();
    const float* qr = &sL[(wave*16 + ln)*XPAD + h*16 + 2*half];
    #pragma unroll
    for (int kk = 0; kk < 4; ++kk) {
      v2f a = *(const v2f*)(qr + kk*4);
      int kcc = kk*4 + 2*half;
      v2f b;
      b.x = ctxL[(h*DKH + ln)*CPAD + kcc];
      b.y = ctxL[(h*DKH + ln)*CPAD + kcc + 1];
      ca[h] = wmma4(a, b, ca[h]);
    }
  }
  #pragma unroll
  for (int h = 0; h < 4; ++h)
    #pragma unroll
    for (int i = 0; i < 8; ++i)
      sL[(wave*16 + i + 8*half)*XPAD + h*16 + ln] = ca[h][i];

  // ---- 4) out = att @ Wr^T + br (pipelined) ----
  v8f co[4] = {vzero8(), vzero8(), vzero8(), vzero8()};
  const float* trow = &sL[(wave*16 + ln)*XPAD + 2*half];
  const float* rrow = &wrL[ln*XPAD + 2*half];
  {
    v2f a_c = *(const v2f*)(trow);
    v2f fb_c[4];
    #pragma unroll
    for (int cg = 0; cg < 4; ++cg)
      fb_c[cg] = *(const v2f*)(rrow + cg*16*XPAD);
    #pragma unroll
    for (int kc = 0; kc < 16; ++kc) {
      v2f a_n = *(const v2f*)(trow + (kc+1)*4);
      v2f fb_n[4];
      #pragma unroll
      for (int cg = 0; cg < 4; ++cg)
        fb_n[cg] = *(const v2f*)(rrow + cg*16*XPAD + (kc+1)*4);
      #pragma unroll
      for (int cg = 0; cg < 4; ++cg)
        co[cg] = wmma4(a_c, fb_c[cg], co[cg]);
      a_c = a_n;
      #pragma unroll
      for (int cg = 0; cg < 4; ++cg) fb_c[cg] = fb_n[cg];
    }
  }
  // stage (with bias) then b128 non-temporal stores: one full 256B row / half-wave
  #pragma unroll
  for (int cg = 0; cg < 4; ++cg) {
    float bb = brL[cg*16+ln];
    #pragma unroll
    for (int i = 0; i < 8; ++i)
      sL[(wave*16 + i + 8*half)*XPAD + cg*16 + ln] = co[cg][i] + bb;
  }
  __syncthreads();
  v4f* op4 = (v4f*)(out + t0*NC);
  for (int i = tid; i < TILE*16; i += THREADS) {
    int r = i >> 4, c4 = (i & 15) << 2;
    v2f p0 = *(const v2f*)&sL[r*XPAD + c4];
    v2f p1 = *(const v2f*)&sL[r*XPAD + c4 + 2];
    v4f o = {p0.x, p0.y, p1.x, p1.y};
    __builtin_nontemporal_store(o, &op4[i]);
  }
}

// ---------------------------------------------------------------------------
extern "C" void kernel_launch(void* const* d_in, const int* in_sizes, int n_in,
                              void* d_out, int out_size, void* d_ws, size_t ws_size,
                              hipStream_t stream) {
  const float* x  = (const float*)d_in[0];
  const float* Wk = (const float*)d_in[1];
  const float* bk = (const float*)d_in[2];
  const float* Wq = (const float*)d_in[3];
  const float* bq = (const float*)d_in[4];
  const float* Wv = (const float*)d_in[5];
  const float* bv = (const float*)d_in[6];
  const float* Wr = (const float*)d_in[7];
  const float* br = (const float*)d_in[8];
  float* out = (float*)d_out;

  float* ctx_part = (float*)d_ws;                       // NSLOT * 2048
  float* se_part  = ctx_part + (size_t)NSLOT*2048;      // NSLOT * 128
  float* ctxn     = se_part  + (size_t)NSLOT*128;       // 2048

  (void)hipMemsetAsync(d_ws, 0, (size_t)NSLOT*(2048 + 128)*sizeof(float), stream);

  const int nblk = (2 * N_PER) / TILE;                  // 4096 workgroups
  size_t lds1 = (size_t)(2*TILE*XPAD + 2*64*XPAD + 192) * sizeof(float);
  size_t lds2 = (size_t)(2*TILE*XPAD + 2*64*XPAD + NH*DKH*CPAD + 128) * sizeof(float);

  ea_pass1<<<nblk, THREADS, lds1, stream>>>(x, Wk, bk, Wv, bv, ctx_part, se_part);
  ea_norm <<<8, 256, 0, stream>>>(ctx_part, se_part, ctxn);
  ea_pass2<<<nblk, THREADS, lds2, stream>>>(x, Wq, bq, Wr, br, ctxn, out);
}